// Net_3624952398105
// MI455X (gfx1250) — compile-verified
//
#include <hip/hip_runtime.h>
#include <math.h>

// MI455X / gfx1250, wave32. Tanh-RNN scan: serial over T=2048, parallel over B=1024.
// One wave per 16-batch tile (64 waves). State kept as S = H^T in WMMA C/D layout;
// per step: C = P (rank-1 input proj, FMA-init) ; C += W_hh x S via 4x
// v_wmma_f32_16x16x4_f32 (A = W_hh constant in regs); H = tanh(C); B-operand for the
// next step rebuilt from registers with v_permlanex16 half-swaps (no LDS round trip).

typedef __attribute__((ext_vector_type(2))) float v2f;
typedef __attribute__((ext_vector_type(4))) float v4f;
typedef __attribute__((ext_vector_type(8))) float v8f;

static constexpr int kT = 2048;
static constexpr int kB = 1024;
static constexpr int kH = 16;

// lane i <-> lane i^16 swap (identity nibble patterns), pure VALU cross-half move.
__device__ __forceinline__ float swap_half(float v) {
  int i = __builtin_bit_cast(int, v);
  int r = __builtin_amdgcn_permlanex16(i, i, 0x76543210, (int)0xFEDCBA98, true, false);
  return __builtin_bit_cast(float, r);
}

__device__ __forceinline__ float fast_tanh(float x) {
#if __has_builtin(__builtin_amdgcn_tanhf)
  return __builtin_amdgcn_tanhf(x);   // v_tanh_f32 (TRANS op on gfx1250)
#else
  return tanhf(x);
#endif
}

__global__ __launch_bounds__(32) void rnn_scan_kernel(
    const float* __restrict__ x,      // [B, T]   (IN_DIM==1)
    const float* __restrict__ h0,     // [1, B, 16]
    const float* __restrict__ W_ih,   // [16, 1]
    const float* __restrict__ W_hh,   // [16, 16]
    const float* __restrict__ b_ih,   // [16]
    const float* __restrict__ b_hh,   // [16]
    const float* __restrict__ W_lin,  // [1, 16]
    const float* __restrict__ b_lin,  // [1]
    float* __restrict__ out)          // [B*T] then [B*16]
{
  const int lane = threadIdx.x;       // 0..31
  const int b    = lane & 15;         // batch row within tile / N column
  const int hh   = lane >> 4;         // half-wave index
  const int b0   = blockIdx.x * 16;

  // ---- constant A operand: W_hh in 16x4 A-layout chunks (lane = m = j, k = 4c+p+2*hh)
  v2f A[4];
#pragma unroll
  for (int c = 0; c < 4; ++c)
    A[c] = *(const v2f*)(W_hh + b * 16 + 4 * c + 2 * hh);

  // ---- per-lane j = v + 8*hh constants (D-layout rows owned by this lane)
  float wih[8], bias[8], wlin[8];
  {
    const int j0 = 8 * hh;
    v4f wa = *(const v4f*)(W_ih  + j0),  wb = *(const v4f*)(W_ih  + j0 + 4);
    v4f ia = *(const v4f*)(b_ih  + j0),  ib = *(const v4f*)(b_ih  + j0 + 4);
    v4f ha = *(const v4f*)(b_hh  + j0),  hb = *(const v4f*)(b_hh  + j0 + 4);
    v4f la = *(const v4f*)(W_lin + j0),  lb = *(const v4f*)(W_lin + j0 + 4);
#pragma unroll
    for (int v = 0; v < 4; ++v) {
      wih[v] = wa[v];  wih[v + 4] = wb[v];
      bias[v] = ia[v] + ha[v];  bias[v + 4] = ib[v] + hb[v];
      wlin[v] = la[v]; wlin[v + 4] = lb[v];
    }
  }
  const float blin = b_lin[0];
  const bool  lowhalf = (hh == 0);

  // ---- state S = H^T in C/D layout: lane (b+16*hh), element v holds H[b][v+8*hh]
  v8f hreg;
  {
    v4f a = *(const v4f*)(h0 + (b0 + b) * 16 + 8 * hh);
    v4f c = *(const v4f*)(h0 + (b0 + b) * 16 + 8 * hh + 4);
#pragma unroll
    for (int v = 0; v < 4; ++v) { hreg[v] = a[v]; hreg[v + 4] = c[v]; }
  }

  const float* xrow = x + (size_t)(b0 + b) * kT;
  float*       orow = out + (size_t)(b0 + b) * kT;

  for (int t4 = 0; t4 < kT / 4; ++t4) {
    v4f xv = *(const v4f*)(xrow + t4 * 4);   // 4 timesteps of input
    v4f oacc;
#pragma unroll
    for (int tt = 0; tt < 4; ++tt) {
      // ---- build B operand (4x16 chunks, lane = n = b, k = p + 2*hh per chunk)
      // chunks 0,1: rows j = 4c+p+2*hh < 8 live in low-half lanes' registers
      // chunks 2,3: rows j >= 8 live in high-half lanes' registers
      v2f Bc[4];
#pragma unroll
      for (int c = 0; c < 2; ++c)
#pragma unroll
        for (int p = 0; p < 2; ++p) {
          float own = hreg[4 * c + p];               // valid on low half (j=4c+p)
          float oth = swap_half(hreg[4 * c + p + 2]);// high half pulls j=4c+p+2 from partner
          Bc[c][p] = lowhalf ? own : oth;
        }
#pragma unroll
      for (int c = 2; c < 4; ++c)
#pragma unroll
        for (int p = 0; p < 2; ++p) {
          float oth = swap_half(hreg[4 * c + p - 8]);// low half pulls j=4c+p from partner
          float own = hreg[4 * c + p - 6];           // valid on high half (j=4c+p+2)
          Bc[c][p] = lowhalf ? oth : own;
        }

      // ---- accumulator init = input projection + biases (rank-1 P in D layout)
      v8f cacc;
#pragma unroll
      for (int v = 0; v < 8; ++v) cacc[v] = fmaf(xv[tt], wih[v], bias[v]);

      // ---- C += W_hh x S : 4 chained f32 WMMAs over K=16
#pragma unroll
      for (int c = 0; c < 4; ++c)
        cacc = __builtin_amdgcn_wmma_f32_16x16x4_f32(
            /*neg_a=*/false, A[c], /*neg_b=*/false, Bc[c],
            /*c_mod=*/(short)0, cacc, /*reuse_a=*/false, /*reuse_b=*/false);

      // ---- nonlinearity
#pragma unroll
      for (int v = 0; v < 8; ++v) hreg[v] = fast_tanh(cacc[v]);

      // ---- fused output projection: out[b,t] = <h_t[b], W_lin> + b_lin
      float o = 0.f;
#pragma unroll
      for (int v = 0; v < 8; ++v) o = fmaf(hreg[v], wlin[v], o);
      o += swap_half(o);                 // combine j=0..7 and j=8..15 partials
      oacc[tt] = o + blin;
    }
    if (lowhalf) *(v4f*)(orow + t4 * 4) = oacc;   // 16B store per 4 steps
  }

  // ---- final hidden state hT[b][j], contiguous 8 floats per lane
  float* hT = out + (size_t)kB * kT + (size_t)(b0 + b) * 16 + 8 * hh;
  *(v4f*)(hT)     = (v4f){hreg[0], hreg[1], hreg[2], hreg[3]};
  *(v4f*)(hT + 4) = (v4f){hreg[4], hreg[5], hreg[6], hreg[7]};
}

extern "C" void kernel_launch(void* const* d_in, const int* in_sizes, int n_in,
                              void* d_out, int out_size, void* d_ws, size_t ws_size,
                              hipStream_t stream) {
  const float* x    = (const float*)d_in[0];
  const float* h0   = (const float*)d_in[1];
  const float* W_ih = (const float*)d_in[2];
  const float* W_hh = (const float*)d_in[3];
  const float* b_ih = (const float*)d_in[4];
  const float* b_hh = (const float*)d_in[5];
  const float* W_ln = (const float*)d_in[6];
  const float* b_ln = (const float*)d_in[7];
  float* out = (float*)d_out;
  (void)in_sizes; (void)n_in; (void)out_size; (void)d_ws; (void)ws_size;

  rnn_scan_kernel<<<kB / 16, 32, 0, stream>>>(x, h0, W_ih, W_hh, b_ih, b_hh,
                                              W_ln, b_ln, out);
}